// OriginalMaxPoolSimpleGCNClassifier_25202868093058
// MI455X (gfx1250) — compile-verified
//
#include <hip/hip_runtime.h>
#include <math.h>

#define HF 128          // hidden/feature width
#define CLS 10          // classes
#define GRAPHS 128      // num_graphs (python scalar; known constant)
#define NEG_SLOPE 0.01f
#define RTPW 4          // row tiles per wave in the GEMM

typedef float v2f __attribute__((ext_vector_type(2)));
typedef float v8f __attribute__((ext_vector_type(8)));

// ------------------------------------------------------------------
// degree / normalization:  deg = 1 (self loop) + in-degree;  dinv = rsqrt(deg)
// dinv is sized/initialized over the padded row count so the GEMM epilogue
// can read it unconditionally.
// ------------------------------------------------------------------
__global__ void k_deg_init(float* __restrict__ deg, int n) {
  int i = blockIdx.x * blockDim.x + threadIdx.x;
  if (i < n) deg[i] = 1.0f;
}

__global__ void k_deg_edges(const int* __restrict__ dst, float* __restrict__ deg, int e) {
  int i = blockIdx.x * blockDim.x + threadIdx.x;
  if (i < e) atomicAdd(&deg[dst[i]], 1.0f);
}

__global__ void k_rsqrt(float* __restrict__ dinv, int n) {
  int i = blockIdx.x * blockDim.x + threadIdx.x;
  if (i < n) {
    float d = dinv[i];
    dinv[i] = d > 0.0f ? rsqrtf(d) : 0.0f;
  }
}

// ------------------------------------------------------------------
// Fused GCN-layer front half:
//   hw   = leaky_slope(A) @ W                    (fp32 WMMA 16x16x4)
//   hout = bias + dinv^2 * hw    (self-loop + bias init; only if writeInit)
//
// One wave owns one 16-wide column strip and RTPW row tiles; it preloads all
// 32 B fragments (64 VGPRs) once and reuses them across row tiles.
// A frag (16x4 f32): lanes 0-15 (K=k0,k0+1), lanes 16-31 (K=k0+2,k0+3), M=lane&15
// B frag (4x16 f32): same K split; N = col0 + (lane&15)
// C (16x16 f32, 8 VGPR): VGPR r -> M=r (lanes 0-15), M=r+8 (lanes 16-31)
// hw/hout are padded to rowTiles*16 rows -> unconditional stores, no exec churn.
// ------------------------------------------------------------------
__global__ __launch_bounds__(128) void k_gemm_wmma(const float* __restrict__ A,
                                                   const float* __restrict__ W,
                                                   float* __restrict__ hw,
                                                   float* __restrict__ hout,
                                                   const float* __restrict__ dinv,
                                                   const float* __restrict__ bias,
                                                   int nrows, int rowTiles,
                                                   float slope, int writeInit) {
  int wave = blockIdx.x * 4 + (threadIdx.x >> 5);
  int lane = threadIdx.x & 31;
  int colTile = wave & 7;                 // 8 column tiles (128/16)
  int rtBase = (wave >> 3) * RTPW;
  if (rtBase >= rowTiles) return;         // wave-uniform
  int col0 = colTile << 4;
  int half = lane >> 4;                   // 0: K pair (0,1); 1: K pair (2,3)
  int l16  = lane & 15;

  // preload the full K=128 strip of B fragments for this column tile
  v2f bf[32];
  #pragma unroll
  for (int i = 0; i < 32; ++i) {
    int ka = i * 4 + half * 2;
    bf[i].x = W[(size_t)ka * HF + col0 + l16];
    bf[i].y = W[(size_t)(ka + 1) * HF + col0 + l16];
  }
  float bcol = bias[col0 + l16];

  int rtEnd = rtBase + RTPW;
  if (rtEnd > rowTiles) rtEnd = rowTiles;

  for (int rt = rtBase; rt < rtEnd; ++rt) {
    int m = rt * 16 + l16;
    const float* arow = A + (size_t)(m < nrows ? m : (nrows - 1)) * HF;

    v8f acc = {};
    #pragma unroll
    for (int i = 0; i < 32; ++i) {
      int ka = i * 4 + half * 2;
      float a0 = arow[ka];
      float a1 = arow[ka + 1];
      v2f a;
      a.x = a0 > 0.0f ? a0 : slope * a0;  // slope==1.0f -> identity (layer 0)
      a.y = a1 > 0.0f ? a1 : slope * a1;
      acc = __builtin_amdgcn_wmma_f32_16x16x4_f32(false, a, false, bf[i], (short)0,
                                                  acc, false, false);
    }

    int baseRow = rt * 16 + half * 8;
    size_t off = (size_t)baseRow * HF + col0 + l16;
    if (writeInit) {
      #pragma unroll
      for (int r = 0; r < 8; ++r) {
        float val = acc[r];
        float dv = dinv[baseRow + r];
        hw[off]   = val;
        hout[off] = fmaf(dv * dv, val, bcol);
        off += HF;
      }
    } else {
      #pragma unroll
      for (int r = 0; r < 8; ++r) {
        hw[off] = acc[r];
        off += HF;
      }
    }
  }
}

// ------------------------------------------------------------------
// fallback (only when ws too small for ping-pong buffers):
// h_out[n] = bias + dinv[n]^2 * hw[n]
// ------------------------------------------------------------------
__global__ void k_self_bias(const float* __restrict__ hw, const float* __restrict__ dinv,
                            const float* __restrict__ bias, float* __restrict__ hout, int n) {
  int idx = blockIdx.x * blockDim.x + threadIdx.x;
  if (idx >= n * 32) return;
  int node = idx >> 5;
  int q = idx & 31;
  float di = dinv[node];
  float nm = di * di;
  float4 v = ((const float4*)hw)[idx];
  float4 b = ((const float4*)bias)[q];
  float4 o;
  o.x = fmaf(nm, v.x, b.x);
  o.y = fmaf(nm, v.y, b.y);
  o.z = fmaf(nm, v.z, b.z);
  o.w = fmaf(nm, v.w, b.w);
  ((float4*)hout)[idx] = o;
}

// ------------------------------------------------------------------
// edge aggregation: h_out[dst] += dinv[src]*dinv[dst] * hw[src]
// one wave per edge, each lane handles 4 contiguous floats (b128 gather, 4 atomics)
// hw (25.6 MB) is L2-resident on MI455X (192 MB L2) -> L2-speed gathers/atomics
// ------------------------------------------------------------------
__global__ void k_edge_scatter(const float* __restrict__ hw, const int* __restrict__ src,
                               const int* __restrict__ dst, const float* __restrict__ dinv,
                               float* __restrict__ hout, int e) {
  int gid = blockIdx.x * blockDim.x + threadIdx.x;
  int edge = gid >> 5;
  int lane = gid & 31;
  if (edge >= e) return;
  int s = src[edge];
  int d = dst[edge];
  float nm = dinv[s] * dinv[d];
  float4 v = ((const float4*)(hw + (size_t)s * HF))[lane];
  float* o = hout + (size_t)d * HF + lane * 4;
  atomicAdd(o + 0, nm * v.x);
  atomicAdd(o + 1, nm * v.y);
  atomicAdd(o + 2, nm * v.z);
  atomicAdd(o + 3, nm * v.w);
}

// ------------------------------------------------------------------
// global max pool via order-independent monotonic-uint atomicMax
// encode: f >= 0 -> bits | 0x80000000 ; f < 0 -> ~bits   (order preserving)
// ------------------------------------------------------------------
__global__ void k_emb_init(unsigned* __restrict__ emb, int n) {
  int i = blockIdx.x * blockDim.x + threadIdx.x;
  if (i < n) emb[i] = 0x007FFFFFu;  // encode(-inf)
}

__global__ void k_pool_max(const float* __restrict__ h, const int* __restrict__ batch,
                           unsigned* __restrict__ emb, int n) {
  int idx = blockIdx.x * blockDim.x + threadIdx.x;
  if (idx >= n * HF) return;
  int node = idx >> 7;
  int f = idx & (HF - 1);
  int g = batch[node];
  unsigned u = __float_as_uint(h[idx]);
  u = (u & 0x80000000u) ? ~u : (u | 0x80000000u);
  atomicMax(&emb[g * HF + f], u);
}

// ------------------------------------------------------------------
// head: decode embeds, z = leaky(emb @ lin_w + lin_b), logits = z @ out_w + out_b, softmax
// one block (128 threads) per graph
// ------------------------------------------------------------------
__global__ __launch_bounds__(HF) void k_head(const unsigned* __restrict__ emb_u,
                                             const float* __restrict__ lin_w,
                                             const float* __restrict__ lin_b,
                                             const float* __restrict__ out_w,
                                             const float* __restrict__ out_b,
                                             float* __restrict__ out_logits,
                                             float* __restrict__ out_probs,
                                             float* __restrict__ out_embeds) {
  __shared__ float e[HF];
  __shared__ float z[HF];
  __shared__ float lg[CLS];
  int g = blockIdx.x;
  int t = threadIdx.x;

  unsigned u = emb_u[g * HF + t];
  unsigned bits = (u & 0x80000000u) ? (u & 0x7FFFFFFFu) : ~u;
  float ef = __uint_as_float(bits);
  e[t] = ef;
  out_embeds[g * HF + t] = ef;
  __syncthreads();

  float acc = lin_b[t];
  #pragma unroll 8
  for (int k = 0; k < HF; ++k) acc = fmaf(e[k], lin_w[k * HF + t], acc);
  z[t] = acc > 0.0f ? acc : NEG_SLOPE * acc;
  __syncthreads();

  if (t < CLS) {
    float a = out_b[t];
    #pragma unroll 8
    for (int k = 0; k < HF; ++k) a = fmaf(z[k], out_w[k * CLS + t], a);
    lg[t] = a;
    out_logits[g * CLS + t] = a;
  }
  __syncthreads();

  if (t == 0) {
    float m = lg[0];
    for (int c = 1; c < CLS; ++c) m = fmaxf(m, lg[c]);
    float ex[CLS];
    float s = 0.0f;
    for (int c = 0; c < CLS; ++c) { ex[c] = expf(lg[c] - m); s += ex[c]; }
    float inv = 1.0f / s;
    for (int c = 0; c < CLS; ++c) out_probs[g * CLS + c] = ex[c] * inv;
  }
}

// ------------------------------------------------------------------
extern "C" void kernel_launch(void* const* d_in, const int* in_sizes, int n_in,
                              void* d_out, int out_size, void* d_ws, size_t ws_size,
                              hipStream_t stream) {
  const float* x       = (const float*)d_in[0];
  const int*   ei      = (const int*)d_in[1];
  const int*   batch   = (const int*)d_in[2];
  // d_in[3] = num_graphs (device scalar); known constant GRAPHS=128
  const float* conv_ws = (const float*)d_in[4];
  const float* conv_bs = (const float*)d_in[5];
  const float* lin_w   = (const float*)d_in[6];
  const float* lin_b   = (const float*)d_in[7];
  const float* out_w   = (const float*)d_in[8];
  const float* out_b   = (const float*)d_in[9];

  const int N = in_sizes[0] / HF;
  const int E = in_sizes[1] / 2;
  const int L = in_sizes[4] / (HF * HF);
  const int* src = ei;
  const int* dst = ei + E;

  const int rowTiles = (N + 15) / 16;
  const int Npad = rowTiles * 16;          // padded row count for unconditional stores
  const size_t fmapElems = (size_t)Npad * HF;
  const size_t dinvElems = ((size_t)(Npad + 255) / 256) * 256;

  // workspace carve-up: dinv | hw | h0 | [h1] | emb_u
  size_t need3 = (dinvElems + 3 * fmapElems + (size_t)GRAPHS * HF) * 4;
  bool fused = ws_size >= need3;           // ping-pong available -> fuse init into GEMM

  float* dinv = (float*)d_ws;
  float* hw   = dinv + dinvElems;
  float* h0   = hw + fmapElems;
  float* h1   = fused ? (h0 + fmapElems) : h0;
  unsigned* emb_u = (unsigned*)((fused ? h1 : h0) + fmapElems);

  float* out_logits = (float*)d_out;
  float* out_probs  = out_logits + GRAPHS * CLS;
  float* out_embeds = out_probs + GRAPHS * CLS;

  const int B = 256;

  // normalization coefficients (over padded rows; pad rows get dinv=1, unused)
  k_deg_init<<<(Npad + B - 1) / B, B, 0, stream>>>(dinv, Npad);
  k_deg_edges<<<(E + B - 1) / B, B, 0, stream>>>(dst, dinv, E);
  k_rsqrt<<<(Npad + B - 1) / B, B, 0, stream>>>(dinv, Npad);

  const int waves = ((rowTiles + RTPW - 1) / RTPW) * 8;
  const int gemmBlocks = (waves + 3) / 4;
  const int nv = Npad * 32;                // float4 elements per feature map

  const float* in = x;
  for (int l = 0; l < L; ++l) {
    // output buffer must differ from input when the init is fused into the GEMM
    float* out = (l == 0) ? h0 : (fused ? ((in == h0) ? h1 : h0) : h0);
    float slope = (l == 0) ? 1.0f : NEG_SLOPE;   // leaky applied on GEMM input read

    k_gemm_wmma<<<gemmBlocks, 128, 0, stream>>>(in, conv_ws + (size_t)l * HF * HF,
                                                hw, out, dinv, conv_bs + (size_t)l * HF,
                                                N, rowTiles, slope, fused ? 1 : 0);
    if (!fused)
      k_self_bias<<<(nv + B - 1) / B, B, 0, stream>>>(hw, dinv,
                                                      conv_bs + (size_t)l * HF, out, Npad);
    k_edge_scatter<<<(E * 32 + B - 1) / B, B, 0, stream>>>(hw, src, dst, dinv, out, E);
    in = out;
  }

  k_emb_init<<<(GRAPHS * HF + B - 1) / B, B, 0, stream>>>(emb_u, GRAPHS * HF);
  k_pool_max<<<(N * HF + B - 1) / B, B, 0, stream>>>(in, batch, emb_u, N);
  k_head<<<GRAPHS, HF, 0, stream>>>(emb_u, lin_w, lin_b, out_w, out_b,
                                    out_logits, out_probs, out_embeds);
}